// LGD_91336774517539
// MI455X (gfx1250) — compile-verified
//
#include <hip/hip_runtime.h>

#define NPTS 8192
#define KNN  16

typedef float v2f __attribute__((ext_vector_type(2)));
typedef float v8f __attribute__((ext_vector_type(8)));

__device__ __forceinline__ v8f wmma_f32(v2f a, v2f b, v8f c) {
    // D = A(16x4,f32) * B(4x16,f32) + C(16x16,f32) on the CDNA5 matrix pipe
    return __builtin_amdgcn_wmma_f32_16x16x4_f32(
        /*neg_a=*/false, a, /*neg_b=*/false, b,
        /*c_mod=*/(short)0, c, /*reuse_a=*/false, /*reuse_b=*/false);
}

__device__ __forceinline__ float prelu(float x, float a) {
    return x >= 0.0f ? x : a * x;
}

// ---------------------------------------------------------------- row norms
__global__ void norms_kernel(const float* __restrict__ X, int ld, int C,
                             float* __restrict__ xx) {
    int i = blockIdx.x * blockDim.x + threadIdx.x;
    if (i < NPTS) {
        float s = 0.0f;
        for (int t = 0; t < C; ++t) { float v = X[(size_t)i * ld + t]; s += v * v; }
        xx[i] = s;
    }
}

// ---------------------------------------------------------------- kNN (top-16)
// 256 thr = 8 waves; block owns 128 rows. A-fragments live in registers for the
// whole sweep; the 64-column B block is staged (zero-padded) into LDS once per
// tile; distances go to LDS; 128 threads keep a register-sorted top-16.
template <int C, int LD>
__launch_bounds__(256)
__global__ void knn_kernel(const float* __restrict__ X,
                           const float* __restrict__ xx,
                           int* __restrict__ idxOut) {
    constexpr int CP   = (C + 3) & ~3;   // K padded to multiple of 4
    constexpr int BST  = CP + 4;         // LDS stride (even)
    __shared__ __align__(16) float Bs[64][BST];
    __shared__ __align__(16) float dt[128][68];

    const int tid = threadIdx.x;
    const int w   = tid >> 5;
    const int ln  = tid & 31;
    const int lo  = ln & 15;
    const int hi  = ln >> 4;
    const int row0 = blockIdx.x * 128;

    // hoist this lane's A fragments into registers (branchless ragged tail)
    v2f areg[CP / 4];
    {
        const float* Arow = X + (size_t)(row0 + w * 16 + lo) * LD;
        #pragma unroll
        for (int t = 0; t < CP / 4; ++t) {
            const int ks = 4 * t + 2 * hi;
            if constexpr (C % 4 == 0 && LD % 2 == 0) {
                areg[t] = *(const v2f*)(Arow + ks);
            } else {
                const int k0c = ks     < C ? ks     : 0;
                const int k1c = ks + 1 < C ? ks + 1 : 0;
                float ax = Arow[k0c], ay = Arow[k1c];
                areg[t].x = (ks     < C) ? ax : 0.0f;
                areg[t].y = (ks + 1 < C) ? ay : 0.0f;
            }
        }
    }

    float d16[16]; int i16[16];
    #pragma unroll
    for (int q = 0; q < 16; ++q) { d16[q] = 3.4e38f; i16[q] = 0; }

    for (int c0 = 0; c0 < NPTS; c0 += 64) {
        // cooperative stage of B block (64 points x CP features, zero padded)
        for (int e = tid; e < 64 * CP; e += 256) {
            const int rr = e / CP, cc = e % CP;
            const int ccc = cc < C ? cc : C - 1;              // clamped, in-bounds
            const float v = X[(size_t)(c0 + rr) * LD + ccc];
            Bs[rr][cc] = (cc < C) ? v : 0.0f;                 // branchless zero-pad
        }
        __syncthreads();

        v8f acc[4] = {};
        #pragma unroll
        for (int t = 0; t < CP / 4; ++t) {
            const int ks = 4 * t + 2 * hi;
            #pragma unroll
            for (int tc = 0; tc < 4; ++tc) {
                const v2f b = *(const v2f*)&Bs[tc * 16 + lo][ks];
                acc[tc] = wmma_f32(areg[t], b, acc[tc]);
            }
        }
        // distances: d = |xi|^2 + |xj|^2 - 2 gram
        #pragma unroll
        for (int tc = 0; tc < 4; ++tc) {
            #pragma unroll
            for (int v = 0; v < 8; ++v) {
                const int rr = w * 16 + v + 8 * hi;
                const int cc = tc * 16 + lo;
                dt[rr][cc] = xx[row0 + rr] + xx[c0 + cc] - 2.0f * acc[tc][v];
            }
        }
        __syncthreads();
        if (tid < 128) {
            for (int cc = 0; cc < 64; ++cc) {
                const float d = dt[tid][cc];
                if (d < d16[15]) {
                    d16[15] = d; i16[15] = c0 + cc;
                    #pragma unroll
                    for (int q = 15; q > 0; --q) {
                        if (d16[q] < d16[q - 1]) {
                            float td = d16[q]; d16[q] = d16[q - 1]; d16[q - 1] = td;
                            int   ti = i16[q]; i16[q] = i16[q - 1]; i16[q - 1] = ti;
                        }
                    }
                }
            }
        }
        __syncthreads();
    }
    if (tid < 128) {
        #pragma unroll
        for (int q = 0; q < 16; ++q) idxOut[(size_t)(row0 + tid) * KNN + q] = i16[q];
    }
}

// ---------------------------------------------------------------- edge features
// E[q][t] = t<Cin ? xj[t]-xi[t] : xi[t-Cin]; lane's (ks,ks+1) pair.
template <int Cin>
__device__ __forceinline__ v2f edge_pair(const float* __restrict__ Xi,
                                         const float* __restrict__ Xj,
                                         int ks /* = 4t + 2hi */, int t4 /* = 4t */) {
    constexpr int C2 = 2 * Cin;
    v2f a;
    if constexpr (Cin % 2 == 0) {
        // pairs never straddle the Cin boundary; boundary test folds per k-step
        if (t4 < Cin) {
            const v2f xj = *(const v2f*)(Xj + ks);
            const v2f xi = *(const v2f*)(Xi + ks);
            a = xj - xi;
        } else {
            a = *(const v2f*)(Xi + (ks - Cin));
        }
    } else {
        const int t0 = ks     < C2 ? ks     : 0;
        const int t1 = ks + 1 < C2 ? ks + 1 : 0;
        const int u0 = t0 < Cin ? t0 : t0 - Cin;
        const int u1 = t1 < Cin ? t1 : t1 - Cin;
        float v0 = (t0 < Cin) ? (Xj[u0] - Xi[u0]) : Xi[u0];
        float v1 = (t1 < Cin) ? (Xj[u1] - Xi[u1]) : Xi[u1];
        a.x = (ks     < C2) ? v0 : 0.0f;
        a.y = (ks + 1 < C2) ? v1 : 0.0f;
    }
    return a;
}

template <int C2>
__device__ __forceinline__ v2f w_pair(const float* __restrict__ W, int ks, int col) {
    v2f b;
    const int k0c = ks     < C2 ? ks     : 0;
    const int k1c = ks + 1 < C2 ? ks + 1 : 0;
    float b0 = W[(size_t)k0c * 64 + col];
    float b1 = W[(size_t)k1c * 64 + col];
    b.x = (ks     < C2) ? b0 : 0.0f;
    b.y = (ks + 1 < C2) ? b1 : 0.0f;
    return b;
}

// ---------------------------------------------------------------- EdgeConv, 2 layers
// One wave per point: E(16 x 2Cin) @ W1 -> prelu -> @ W2 -> prelu -> max over 16.
template <int Cin, int LD>
__launch_bounds__(256)
__global__ void edgeconv2_kernel(const float* __restrict__ Xin,
                                 const int* __restrict__ nbrs,
                                 const float* __restrict__ W1, const float* __restrict__ a1p,
                                 const float* __restrict__ W2, const float* __restrict__ a2p,
                                 float* __restrict__ Xout, int oOff) {
    constexpr int C2  = 2 * Cin;
    constexpr int C2P = (C2 + 3) & ~3;
    __shared__ __align__(16) float f1s[8][16][66];   // layer-1 activations, re-layout
    __shared__ __align__(16) float W2t[64][66];      // W2 transposed: W2t[col][k]

    const int tid = threadIdx.x;
    const int w   = tid >> 5;
    const int ln  = tid & 31;
    const int lo  = ln & 15;
    const int hi  = ln >> 4;
    const int i   = blockIdx.x * 8 + w;
    const float a1 = *a1p, a2 = *a2p;

    // stage W2^T (shared by the 8 waves)
    for (int e = tid; e < 64 * 64; e += 256) {
        const int c = e & 63, k = e >> 6;
        W2t[c][k] = W2[(size_t)k * 64 + c];
    }

    const int nbr = nbrs[(size_t)i * KNN + lo];
    const float* Xi = Xin + (size_t)i   * LD;
    const float* Xj = Xin + (size_t)nbr * LD;

    // layer 1
    v8f acc[4] = {};
    #pragma unroll
    for (int t = 0; t < C2P / 4; ++t) {
        const int ks = 4 * t + 2 * hi;
        const v2f a = edge_pair<Cin>(Xi, Xj, ks, 4 * t);
        #pragma unroll
        for (int nt = 0; nt < 4; ++nt) {
            const v2f b = w_pair<C2>(W1, ks, nt * 16 + lo);
            acc[nt] = wmma_f32(a, b, acc[nt]);
        }
    }
    #pragma unroll
    for (int nt = 0; nt < 4; ++nt)
        #pragma unroll
        for (int v = 0; v < 8; ++v)
            f1s[w][v + 8 * hi][nt * 16 + lo] = prelu(acc[nt][v], a1);
    __syncthreads();   // orders W2t staging and f1s stores before reads

    // layer 2 (K = 64)
    v8f acc2[4] = {};
    #pragma unroll
    for (int t = 0; t < 16; ++t) {
        const int ks = 4 * t + 2 * hi;
        const v2f a = *(const v2f*)&f1s[w][lo][ks];
        #pragma unroll
        for (int nt = 0; nt < 4; ++nt) {
            const v2f b = *(const v2f*)&W2t[nt * 16 + lo][ks];
            acc2[nt] = wmma_f32(a, b, acc2[nt]);
        }
    }
    // prelu + max over the 16 neighbor rows
    #pragma unroll
    for (int nt = 0; nt < 4; ++nt) {
        float m = -3.4e38f;
        #pragma unroll
        for (int v = 0; v < 8; ++v) m = fmaxf(m, prelu(acc2[nt][v], a2));
        m = fmaxf(m, __shfl_xor(m, 16, 32));
        if (hi == 0) Xout[(size_t)i * 192 + oOff + nt * 16 + lo] = m;
    }
}

// ---------------------------------------------------------------- EdgeConv, 1 layer
template <int Cin, int LD>
__launch_bounds__(256)
__global__ void edgeconv1_kernel(const float* __restrict__ Xin,
                                 const int* __restrict__ nbrs,
                                 const float* __restrict__ W1, const float* __restrict__ a1p,
                                 float* __restrict__ Xout, int oOff) {
    constexpr int C2  = 2 * Cin;
    constexpr int C2P = (C2 + 3) & ~3;
    const int tid = threadIdx.x;
    const int w   = tid >> 5;
    const int ln  = tid & 31;
    const int lo  = ln & 15;
    const int hi  = ln >> 4;
    const int i   = blockIdx.x * 8 + w;
    const float a1 = *a1p;

    const int nbr = nbrs[(size_t)i * KNN + lo];
    const float* Xi = Xin + (size_t)i   * LD;
    const float* Xj = Xin + (size_t)nbr * LD;

    v8f acc[4] = {};
    #pragma unroll
    for (int t = 0; t < C2P / 4; ++t) {
        const int ks = 4 * t + 2 * hi;
        const v2f a = edge_pair<Cin>(Xi, Xj, ks, 4 * t);
        #pragma unroll
        for (int nt = 0; nt < 4; ++nt) {
            const v2f b = w_pair<C2>(W1, ks, nt * 16 + lo);
            acc[nt] = wmma_f32(a, b, acc[nt]);
        }
    }
    #pragma unroll
    for (int nt = 0; nt < 4; ++nt) {
        float m = -3.4e38f;
        #pragma unroll
        for (int v = 0; v < 8; ++v) m = fmaxf(m, prelu(acc[nt][v], a1));
        m = fmaxf(m, __shfl_xor(m, 16, 32));
        if (hi == 0) Xout[(size_t)i * 192 + oOff + nt * 16 + lo] = m;
    }
}

// ---------------------------------------------------------------- LDS-tiled WMMA GEMM
// C[M,Nc] = prelu(A @ B + bias); MAXMODE: column-max (after prelu) via
// monotonic-uint atomics instead of a store. Tile: 128 rows x 64 cols, K chunks
// of 64 staged transposed in LDS.
template <int Kd, bool MAXMODE>
__launch_bounds__(256)
__global__ void gemm_prelu_kernel(const float* __restrict__ A, int lda,
                                  const float* __restrict__ B, int ldb,
                                  const float* __restrict__ bias,
                                  const float* __restrict__ alphap,
                                  float* __restrict__ Cout, int ldc,
                                  unsigned* __restrict__ maxOut) {
    __shared__ __align__(16) float Bt[64][68];       // Bt[col][k]
    const int tid = threadIdx.x;
    const int w   = tid >> 5;
    const int ln  = tid & 31;
    const int lo  = ln & 15;
    const int hi  = ln >> 4;
    const int row0 = blockIdx.x * 128 + w * 16;
    const int col0 = blockIdx.y * 64;
    const float alpha = *alphap;

    v8f acc[4] = {};
    const float* Arow = A + (size_t)(row0 + lo) * lda;
    for (int kb = 0; kb < Kd; kb += 64) {
        for (int e = tid; e < 64 * 64; e += 256) {
            const int c = e & 63, kk = e >> 6;       // consecutive c -> coalesced
            Bt[c][kk] = B[(size_t)(kb + kk) * ldb + col0 + c];
        }
        __syncthreads();
        #pragma unroll
        for (int t = 0; t < 16; ++t) {
            const int ksl = 4 * t + 2 * hi;
            const v2f a = *(const v2f*)(Arow + kb + ksl);
            #pragma unroll
            for (int nt = 0; nt < 4; ++nt) {
                const v2f b = *(const v2f*)&Bt[nt * 16 + lo][ksl];
                acc[nt] = wmma_f32(a, b, acc[nt]);
            }
        }
        __syncthreads();
    }
    if constexpr (MAXMODE) {
        #pragma unroll
        for (int nt = 0; nt < 4; ++nt) {
            float m = -3.4e38f;
            #pragma unroll
            for (int v = 0; v < 8; ++v) m = fmaxf(m, prelu(acc[nt][v], alpha));
            m = fmaxf(m, __shfl_xor(m, 16, 32));
            if (hi == 0) {
                unsigned u = __float_as_uint(m);
                u = (u & 0x80000000u) ? ~u : (u | 0x80000000u);
                atomicMax(&maxOut[col0 + nt * 16 + lo], u);
            }
        }
    } else {
        #pragma unroll
        for (int nt = 0; nt < 4; ++nt) {
            const int col = col0 + nt * 16 + lo;
            const float bi = bias ? bias[col] : 0.0f;
            #pragma unroll
            for (int v = 0; v < 8; ++v) {
                const float f = prelu(acc[nt][v] + bi, alpha);
                Cout[(size_t)(row0 + v + 8 * hi) * ldc + col] = f;
            }
        }
    }
}

// ------------------------------------------- bias = x5max @ L1[192:1216, :]
__global__ void bias_from_x5_kernel(const unsigned* __restrict__ x5enc,
                                    const float* __restrict__ L1,
                                    float* __restrict__ bias) {
    const int c = threadIdx.x;            // 256 threads, 1 block
    float s = 0.0f;
    for (int k = 0; k < 1024; ++k) {
        const unsigned e = x5enc[k];
        const float f = (e & 0x80000000u) ? __uint_as_float(e & 0x7FFFFFFFu)
                                          : __uint_as_float(~e);
        s += f * L1[(size_t)(192 + k) * 256 + c];
    }
    bias[c] = s;
}

// ---------------------------------------------------------------- final L4 (128 -> 4)
__global__ void l4_kernel(const float* __restrict__ H, const float* __restrict__ W,
                          const float* __restrict__ ap, float* __restrict__ out) {
    const int i = blockIdx.x * blockDim.x + threadIdx.x;
    if (i < NPTS) {
        const float a = *ap;
        float s0 = 0.f, s1 = 0.f, s2 = 0.f, s3 = 0.f;
        for (int k = 0; k < 128; ++k) {
            const float h = H[(size_t)i * 128 + k];
            s0 += h * W[k * 4 + 0]; s1 += h * W[k * 4 + 1];
            s2 += h * W[k * 4 + 2]; s3 += h * W[k * 4 + 3];
        }
        out[(size_t)i * 4 + 0] = prelu(s0, a);
        out[(size_t)i * 4 + 1] = prelu(s1, a);
        out[(size_t)i * 4 + 2] = prelu(s2, a);
        out[(size_t)i * 4 + 3] = prelu(s3, a);
    }
}

extern "C" void kernel_launch(void* const* d_in, const int* in_sizes, int n_in,
                              void* d_out, int out_size, void* d_ws, size_t ws_size,
                              hipStream_t stream) {
    (void)in_sizes; (void)n_in; (void)out_size; (void)ws_size;
    const float* x   = (const float*)d_in[0];
    const float* W1a = (const float*)d_in[1];  const float* a1a = (const float*)d_in[2];
    const float* W1b = (const float*)d_in[3];  const float* a1b = (const float*)d_in[4];
    const float* W2a = (const float*)d_in[5];  const float* a2a = (const float*)d_in[6];
    const float* W2b = (const float*)d_in[7];  const float* a2b = (const float*)d_in[8];
    const float* W3  = (const float*)d_in[9];  const float* a3  = (const float*)d_in[10];
    const float* W4  = (const float*)d_in[11]; const float* a4  = (const float*)d_in[12];
    const float* L1  = (const float*)d_in[13]; const float* aL1 = (const float*)d_in[14];
    const float* L2  = (const float*)d_in[15]; const float* aL2 = (const float*)d_in[16];
    const float* L3  = (const float*)d_in[17]; const float* aL3 = (const float*)d_in[18];
    const float* L4w = (const float*)d_in[19]; const float* aL4 = (const float*)d_in[20];

    float* wsf = (float*)d_ws;
    float*    xx    = wsf;                                   // N
    float*    X123  = wsf + NPTS;                            // N x 192 (x1|x2|x3)
    int*      idx   = (int*)(X123 + (size_t)NPTS * 192);     // N x 16, reused
    unsigned* x5    = (unsigned*)(idx + (size_t)NPTS * KNN); // 1024 (encoded)
    float*    bias1 = (float*)(x5 + 1024);                   // 256
    float*    h1    = bias1 + 256;                           // N x 256
    float*    h2    = h1 + (size_t)NPTS * 256;               // N x 256

    // ---- stage 1: knn(x) -> edgeconv -> x1
    norms_kernel<<<NPTS / 256, 256, 0, stream>>>(x, 9, 9, xx);
    knn_kernel<9, 9><<<NPTS / 128, 256, 0, stream>>>(x, xx, idx);
    edgeconv2_kernel<9, 9><<<NPTS / 8, 256, 0, stream>>>(x, idx, W1a, a1a, W1b, a1b,
                                                         X123, 0);
    // ---- stage 2: knn(x1) -> edgeconv -> x2
    norms_kernel<<<NPTS / 256, 256, 0, stream>>>(X123, 192, 64, xx);
    knn_kernel<64, 192><<<NPTS / 128, 256, 0, stream>>>(X123, xx, idx);
    edgeconv2_kernel<64, 192><<<NPTS / 8, 256, 0, stream>>>(X123, idx, W2a, a2a, W2b, a2b,
                                                            X123, 64);
    // ---- stage 3: knn(x2) -> edgeconv -> x3
    norms_kernel<<<NPTS / 256, 256, 0, stream>>>(X123 + 64, 192, 64, xx);
    knn_kernel<64, 192><<<NPTS / 128, 256, 0, stream>>>(X123 + 64, xx, idx);
    edgeconv1_kernel<64, 192><<<NPTS / 8, 256, 0, stream>>>(X123 + 64, idx, W3, a3,
                                                            X123, 128);
    // ---- x5 = colmax(prelu(X123 @ W4))  (encoded atomic max)
    hipMemsetAsync(x5, 0, 1024 * sizeof(unsigned), stream);
    gemm_prelu_kernel<192, true><<<dim3(NPTS / 128, 1024 / 64), 256, 0, stream>>>(
        X123, 192, W4, 1024, nullptr, a4, nullptr, 0, x5);
    // ---- fold broadcast x5 into L1 bias: bias1 = x5 @ L1[192:,:]
    bias_from_x5_kernel<<<1, 256, 0, stream>>>(x5, L1, bias1);
    // ---- pointwise MLP
    gemm_prelu_kernel<192, false><<<dim3(NPTS / 128, 256 / 64), 256, 0, stream>>>(
        X123, 192, L1, 256, bias1, aL1, h1, 256, nullptr);
    gemm_prelu_kernel<256, false><<<dim3(NPTS / 128, 256 / 64), 256, 0, stream>>>(
        h1, 256, L2, 256, nullptr, aL2, h2, 256, nullptr);
    gemm_prelu_kernel<256, false><<<dim3(NPTS / 128, 128 / 64), 256, 0, stream>>>(
        h2, 256, L3, 128, nullptr, aL3, h1, 128, nullptr);
    l4_kernel<<<NPTS / 256, 256, 0, stream>>>(h1, L4w, aL4, (float*)d_out);
}